// NewFusion_33371895890354
// MI455X (gfx1250) — compile-verified
//
#include <hip/hip_runtime.h>
#include <hip/hip_bf16.h>

typedef float v2f __attribute__((ext_vector_type(2)));
typedef float v8f __attribute__((ext_vector_type(8)));

#define HW   4096
#define NPIX 16384   // B*H*W = 4*64*64

__device__ __forceinline__ v8f wmma_f32(v2f a, v2f b, v8f c) {
  // V_WMMA_F32_16X16X4_F32 : D(16x16 f32) = A(16x4 f32) * B(4x16 f32) + C
  return __builtin_amdgcn_wmma_f32_16x16x4_f32(false, a, false, b, (short)0, c,
                                               false, false);
}

__device__ __forceinline__ float hswish_f(float x) {
  return x * fminf(fmaxf(x + 3.0f, 0.0f), 6.0f) * (1.0f / 6.0f);
}

// ---------------------------------------------------------------------------
// Generic fused 1x1-conv GEMM:  out[b,co,p] = epil( sum_k W[co,k]*X[b,cin(k),p] )
//   input may be a channel-concat of (in0: split ch) and (in1: Cin-split ch)
//   grouped convs: group g = co/gso, input channel = g*gsi + k, k < gsi
//   mode: 0=none  1=BN(y=x*g+b)  2=BN+hswish  3=bias+hswish
// grid: (NPIX/512, Cout/16), block: 256 = 8 waves, all in N.
// wave tile: M=16 x N=64 (4 accumulators); B loads unique per wave,
// A (weights) shared across waves via L0.
// ---------------------------------------------------------------------------
__global__ __launch_bounds__(256) void gemm1x1_kernel(
    const float* __restrict__ in0, const float* __restrict__ in1, int split,
    const float* __restrict__ W, float* __restrict__ out,
    int Cout, int Cin, int gsi, int gso,
    const float* __restrict__ gamma, const float* __restrict__ beta, int mode) {
  const int lane = threadIdx.x & 31;
  const int wave = threadIdx.x >> 5;
  const int l15 = lane & 15;
  const int hi = (lane >= 16) ? 1 : 0;

  const int m0 = blockIdx.y * 16;                  // output-channel tile base
  const int n0 = blockIdx.x * 512 + wave * 64;     // global pixel base
  const int bb = n0 >> 12;                         // batch
  const int p0 = n0 & 4095;                        // pixel within image

  const int row = m0 + l15;                        // this lane's A row (co)
  const int g = m0 / gso;                          // conv group
  const int kbaseg = g * gsi;                      // group's input-ch base

  const float* Wrow = W + (size_t)row * gsi + (hi ? 2 : 0);

  // concat split point inside this group's K range
  int seg0 = split - kbaseg;
  if (seg0 < 0) seg0 = 0;
  if (seg0 > gsi) seg0 = gsi;

  v8f acc[4] = {v8f{}, v8f{}, v8f{}, v8f{}};

  // segment 0: channels from in0
  {
    const float* bp = in0 + ((size_t)bb * split + kbaseg + (hi ? 2 : 0)) * HW +
                      p0 + l15;
    for (int k0 = 0; k0 < seg0; k0 += 4) {
      v2f a = *(const v2f*)(Wrow + k0);
      __builtin_prefetch(bp + 8 * (size_t)HW, 0, 0);  // speculative GL2 pf
#pragma unroll
      for (int s = 0; s < 4; ++s) {
        v2f b;
        b.x = bp[s * 16];        // K = k0   (+2 for hi lanes)
        b.y = bp[HW + s * 16];   // K = k0+1 (+2 for hi lanes)
        acc[s] = wmma_f32(a, b, acc[s]);
      }
      bp += 4 * (size_t)HW;
    }
  }
  // segment 1: channels from in1
  if (seg0 < gsi) {
    const float* bp = in1 +
                      ((size_t)bb * (Cin - split) + (kbaseg + seg0 - split) +
                       (hi ? 2 : 0)) * HW + p0 + l15;
    for (int k0 = seg0; k0 < gsi; k0 += 4) {
      v2f a = *(const v2f*)(Wrow + k0);
      __builtin_prefetch(bp + 8 * (size_t)HW, 0, 0);
#pragma unroll
      for (int s = 0; s < 4; ++s) {
        v2f b;
        b.x = bp[s * 16];
        b.y = bp[HW + s * 16];
        acc[s] = wmma_f32(a, b, acc[s]);
      }
      bp += 4 * (size_t)HW;
    }
  }

  // epilogue + store; C/D layout: VGPR v -> M=v (lanes 0-15) / M=v+8 (16-31)
#pragma unroll
  for (int v = 0; v < 8; ++v) {
    int co = m0 + v + (hi ? 8 : 0);
    float g1 = 1.0f, b1 = 0.0f;
    if (mode == 1 || mode == 2) { g1 = gamma[co]; b1 = beta[co]; }
    else if (mode == 3)         { b1 = beta[co]; }
    float* op = out + ((size_t)bb * Cout + co) * HW + p0 + l15;
#pragma unroll
    for (int s = 0; s < 4; ++s) {
      float y = fmaf(acc[s][v], g1, b1);
      if (mode >= 2) y = hswish_f(y);
      op[s * 16] = y;
    }
  }
}

// ---------------------------------------------------------------------------
// Depthwise KSxKS conv (templated so taps fully unroll), fused epilogue.
// ---------------------------------------------------------------------------
template <int KS, int PAD>
__global__ __launch_bounds__(256) void dwconv_kernel(
    const float* __restrict__ in, const float* __restrict__ w,
    float* __restrict__ out, int C,
    const float* __restrict__ gamma, const float* __restrict__ beta, int mode) {
  size_t idx = (size_t)blockIdx.x * 256 + threadIdx.x;
  int p = (int)(idx & 4095);
  int plane = (int)(idx >> 12);          // b*C + c
  int c = plane % C;
  int x = p & 63, y = p >> 6;
  const float* ip = in + ((size_t)plane << 12);
  const float* wp = w + (size_t)c * KS * KS;

  float acc = 0.0f;
#pragma unroll
  for (int ky = 0; ky < KS; ++ky) {
    int iy = y + ky - PAD;
    if ((unsigned)iy >= 64u) continue;
    const float* rp = ip + iy * 64;
#pragma unroll
    for (int kx = 0; kx < KS; ++kx) {
      int ix = x + kx - PAD;
      if ((unsigned)ix >= 64u) continue;
      acc = fmaf(rp[ix], wp[ky * KS + kx], acc);
    }
  }
  float g1 = 1.0f, b1 = 0.0f;
  if (mode == 1 || mode == 2) { g1 = gamma[c]; b1 = beta[c]; }
  else if (mode == 3)         { b1 = beta[c]; }
  float yv = fmaf(acc, g1, b1);
  if (mode >= 2) yv = hswish_f(yv);
  out[idx] = yv;
}

// ---------------------------------------------------------------------------
// Attention phase 1: ctx[d,e] = sum_n relu(k[n,d]) * vpad[n,e]
//   d in [0,32), e in [0,33) (e==32 -> vpad==1).
// one block per (att a, batch b, head hh): 128 blocks, 8 waves.
// Each wave accumulates over 512 pixels via WMMA, partial 32x48 tiles
// reduced across waves through LDS.
// ---------------------------------------------------------------------------
__global__ __launch_bounds__(256) void att_ctx_kernel(
    const float* __restrict__ kvbuf, const float* __restrict__ kvsbuf,
    float* __restrict__ ctxg) {
  const int id = blockIdx.x;
  const int hh = id & 15;
  const int b = (id >> 4) & 3;
  const int a = id >> 6;

  const int lane = threadIdx.x & 31;
  const int wave = threadIdx.x >> 5;
  const int l15 = lane & 15;
  const int hi = (lane >= 16) ? 1 : 0;

  const float* part = (hh < 8) ? kvbuf : kvsbuf;
  const int cb = a * 512 + (hh & 7) * 64;
  const float* kbase = part + ((size_t)b * 1024 + cb) * HW;
  const float* vbase = kbase + (size_t)32 * HW;

  // constant B fragment for e-tile 2 (e = 32 + l15): vpad column 32 == 1.0
  v2f b2;
  b2.x = (l15 == 0) ? 1.0f : 0.0f;
  b2.y = b2.x;

  v8f acc[6] = {v8f{}, v8f{}, v8f{}, v8f{}, v8f{}, v8f{}};

  const int nbeg = wave * 512, nend = nbeg + 512;
  for (int n0 = nbeg; n0 < nend; n0 += 4) {
    const int pk = n0 + (hi ? 2 : 0);
    v2f af[2];
#pragma unroll
    for (int dt = 0; dt < 2; ++dt) {
      const float* kp = kbase + (size_t)(dt * 16 + l15) * HW + pk;
      af[dt].x = fmaxf(kp[0], 0.0f);
      af[dt].y = fmaxf(kp[1], 0.0f);
    }
    v2f bf[2];
#pragma unroll
    for (int et = 0; et < 2; ++et) {
      const float* vp = vbase + (size_t)(et * 16 + l15) * HW + pk;
      bf[et].x = vp[0];
      bf[et].y = vp[1];
    }
#pragma unroll
    for (int dt = 0; dt < 2; ++dt) {
      acc[dt * 3 + 0] = wmma_f32(af[dt], bf[0], acc[dt * 3 + 0]);
      acc[dt * 3 + 1] = wmma_f32(af[dt], bf[1], acc[dt * 3 + 1]);
      acc[dt * 3 + 2] = wmma_f32(af[dt], b2,    acc[dt * 3 + 2]);
    }
  }

  // cross-wave reduction through LDS
  __shared__ float red[8 * 1536];
#pragma unroll
  for (int ai = 0; ai < 6; ++ai)
#pragma unroll
    for (int v = 0; v < 8; ++v)
      red[wave * 1536 + ai * 256 + v * 32 + lane] = acc[ai][v];
  __syncthreads();

  const size_t cbase = (size_t)((a * 4 + b) * 16 + hh) * (32 * 33);
  for (int idx = threadIdx.x; idx < 1536; idx += 256) {
    float s = 0.0f;
#pragma unroll
    for (int w8 = 0; w8 < 8; ++w8) s += red[w8 * 1536 + idx];
    int ai = idx >> 8;
    int v = (idx >> 5) & 7;
    int l = idx & 31;
    int dt = ai / 3, et = ai % 3;
    int d = dt * 16 + v + ((l < 16) ? 0 : 8);
    int e = et * 16 + (l & 15);
    if (e < 33) ctxg[cbase + d * 33 + e] = s;
  }
}

// ---------------------------------------------------------------------------
// Attention phase 2: out[n,j] = (sum_d relu(q[n,d])*ctx[d,j]) / (den(n)+eps),
//   den(n) = sum_d relu(q[n,d])*ctx[d,32].
// grid: 512 blocks = 4 pixel-chunks x (a,b,hh). A = qf (M=pixels), B = ctx(LDS).
// ---------------------------------------------------------------------------
__global__ __launch_bounds__(256) void att_out_kernel(
    const float* __restrict__ qt, const float* __restrict__ qts,
    const float* __restrict__ qrgb, const float* __restrict__ qrgbs,
    const float* __restrict__ ctxg,
    float* __restrict__ out1, float* __restrict__ out2) {
  const int chunk = blockIdx.x & 3;
  const int id = blockIdx.x >> 2;
  const int hh = id & 15;
  const int b = (id >> 4) & 3;
  const int a = id >> 6;

  const int lane = threadIdx.x & 31;
  const int wave = threadIdx.x >> 5;
  const int l15 = lane & 15;
  const int hi = (lane >= 16) ? 1 : 0;

  __shared__ float sctx[32 * 33];
  const size_t cbase = (size_t)((a * 4 + b) * 16 + hh) * (32 * 33);
  for (int i = threadIdx.x; i < 32 * 33; i += 256) sctx[i] = ctxg[cbase + i];
  __syncthreads();

  const float* qpart = (a == 0) ? ((hh < 8) ? qt : qts)
                                : ((hh < 8) ? qrgb : qrgbs);
  const float* qbase = qpart + ((size_t)b * 256 + (hh & 7) * 32) * HW;
  float* obase = ((a == 0) ? out1 : out2) + ((size_t)b * 512 + hh * 32) * HW;

  const int nbeg = chunk * 1024 + wave * 128;
  for (int n0 = nbeg; n0 < nbeg + 128; n0 += 16) {
    v8f acc[3] = {v8f{}, v8f{}, v8f{}};
    const int p = n0 + l15;                         // this lane's A-row pixel
#pragma unroll
    for (int ks = 0; ks < 32; ks += 4) {
      const int kb = ks + (hi ? 2 : 0);
      v2f af;
      af.x = fmaxf(qbase[(size_t)kb * HW + p], 0.0f);
      af.y = fmaxf(qbase[(size_t)(kb + 1) * HW + p], 0.0f);
#pragma unroll
      for (int et = 0; et < 3; ++et) {
        int e = et * 16 + l15;
        v2f bf;
        bf.x = (e < 33) ? sctx[kb * 33 + e] : 0.0f;
        bf.y = (e < 33) ? sctx[(kb + 1) * 33 + e] : 0.0f;
        acc[et] = wmma_f32(af, bf, acc[et]);
      }
    }
#pragma unroll
    for (int v = 0; v < 8; ++v) {
      int pix = n0 + v + (hi ? 8 : 0);              // this element's M row
      float den = __shfl(acc[2][v], hi ? 16 : 0, 32);
      float dinv = 1.0f / (den + 1e-15f);
#pragma unroll
      for (int et = 0; et < 2; ++et) {
        int j = et * 16 + l15;                      // output channel in head
        obase[(size_t)j * HW + pix] = acc[et][v] * dinv;
      }
    }
  }
}

// ---------------------------------------------------------------------------
extern "C" void kernel_launch(void* const* d_in, const int* in_sizes, int n_in,
                              void* d_out, int out_size, void* d_ws,
                              size_t ws_size, hipStream_t stream) {
  const float* rgb = (const float*)d_in[0];
  const float* t = (const float*)d_in[1];
  const float* m1_inv_w = (const float*)d_in[2];
  const float* m1_inv_g = (const float*)d_in[3];
  const float* m1_inv_b = (const float*)d_in[4];
  const float* m1_dw_w = (const float*)d_in[5];
  const float* m1_dw_g = (const float*)d_in[6];
  const float* m1_dw_b = (const float*)d_in[7];
  const float* m1_pw_w = (const float*)d_in[8];
  const float* m1_pw_g = (const float*)d_in[9];
  const float* m1_pw_b = (const float*)d_in[10];
  const float* kv_w = (const float*)d_in[11];
  const float* qrgb_w = (const float*)d_in[12];
  const float* qt_w = (const float*)d_in[13];
  const float* agg_kv_dw = (const float*)d_in[14];
  const float* agg_kv_pw = (const float*)d_in[15];
  const float* agg_t_dw = (const float*)d_in[16];
  const float* agg_t_pw = (const float*)d_in[17];
  const float* agg_rgb_dw = (const float*)d_in[18];
  const float* agg_rgb_pw = (const float*)d_in[19];
  const float* proj_w = (const float*)d_in[20];
  const float* m2_inv_w = (const float*)d_in[21];
  const float* m2_inv_bias = (const float*)d_in[22];
  const float* m2_dw_w = (const float*)d_in[23];
  const float* m2_dw_bias = (const float*)d_in[24];
  const float* m2_pw_w = (const float*)d_in[25];
  const float* m2_pw_g = (const float*)d_in[26];
  const float* m2_pw_b = (const float*)d_in[27];

  float* ws = (float*)d_ws;
  const size_t S256 = (size_t)NPIX * 256;
  const size_t S512 = (size_t)NPIX * 512;
  const size_t S1024 = (size_t)NPIX * 1024;

  float* x1 = ws;                    // m1_inv out -> rgbt -> proj out
  float* x2 = x1 + S256;             // m1_dw out  -> q_t
  float* kvb = x2 + S256;            // kv  (later m2_dw out)
  float* qrgbb = kvb + S1024;        // q_rgb
  float* kvdw = qrgbb + S256;        // kv after dw5 (later out1/out2)
  float* kvs = kvdw + S1024;         // kv_s (later m2_inv out)
  float* qtdw = kvs + S1024;         // qt/qrgb dw5 scratch (reused)
  float* qtsb = qtdw + S256;         // qt_s
  float* qrgbsb = qtsb + S256;       // qrgb_s
  float* ctxb = qrgbsb + S256;       // 2*4*16*32*33 floats
  float* out1 = kvdw;
  float* out2 = kvdw + S512;
  float* m2a = kvs;
  float* m2b = kvb;

  const dim3 blk(256);
  const dim3 g256(NPIX / 512, 256 / 16);   // Cout=256
  const dim3 g1024(NPIX / 512, 1024 / 16); // Cout=1024

  // m1: inverted-bottleneck on concat(rgb,t)
  gemm1x1_kernel<<<g256, blk, 0, stream>>>(rgb, t, 256, m1_inv_w, x1, 256, 512,
                                           512, 256, m1_inv_g, m1_inv_b, 2);
  dwconv_kernel<3, 1><<<(4 * 256 * HW) / 256, blk, 0, stream>>>(
      x1, m1_dw_w, x2, 256, m1_dw_g, m1_dw_b, 2);
  gemm1x1_kernel<<<g256, blk, 0, stream>>>(x2, x2, 256, m1_pw_w, x1, 256, 256,
                                           256, 256, m1_pw_g, m1_pw_b, 1);
  // projections
  gemm1x1_kernel<<<g1024, blk, 0, stream>>>(x1, x1, 256, kv_w, kvb, 1024, 256,
                                            256, 1024, nullptr, nullptr, 0);
  gemm1x1_kernel<<<g256, blk, 0, stream>>>(t, t, 256, qt_w, x2, 256, 256, 256,
                                           256, nullptr, nullptr, 0);
  gemm1x1_kernel<<<g256, blk, 0, stream>>>(rgb, rgb, 256, qrgb_w, qrgbb, 256,
                                           256, 256, 256, nullptr, nullptr, 0);
  // aggregation branches: dw5x5 + grouped 1x1 (32ch groups)
  dwconv_kernel<5, 2><<<(4 * 1024 * HW) / 256, blk, 0, stream>>>(
      kvb, agg_kv_dw, kvdw, 1024, nullptr, nullptr, 0);
  gemm1x1_kernel<<<g1024, blk, 0, stream>>>(kvdw, kvdw, 1024, agg_kv_pw, kvs,
                                            1024, 1024, 32, 32, nullptr,
                                            nullptr, 0);
  dwconv_kernel<5, 2><<<(4 * 256 * HW) / 256, blk, 0, stream>>>(
      x2, agg_t_dw, qtdw, 256, nullptr, nullptr, 0);
  gemm1x1_kernel<<<g256, blk, 0, stream>>>(qtdw, qtdw, 256, agg_t_pw, qtsb, 256,
                                           256, 32, 32, nullptr, nullptr, 0);
  dwconv_kernel<5, 2><<<(4 * 256 * HW) / 256, blk, 0, stream>>>(
      qrgbb, agg_rgb_dw, qtdw, 256, nullptr, nullptr, 0);
  gemm1x1_kernel<<<g256, blk, 0, stream>>>(qtdw, qtdw, 256, agg_rgb_pw, qrgbsb,
                                           256, 256, 32, 32, nullptr, nullptr,
                                           0);
  // relu linear attention (both streams)
  att_ctx_kernel<<<128, blk, 0, stream>>>(kvb, kvs, ctxb);
  att_out_kernel<<<512, blk, 0, stream>>>(x2, qtsb, qrgbb, qrgbsb, ctxb, out1,
                                          out2);
  // projection of concat(out1,out2) + m2 inverted bottleneck
  gemm1x1_kernel<<<g256, blk, 0, stream>>>(out1, out2, 512, proj_w, x1, 256,
                                           1024, 1024, 256, nullptr, nullptr,
                                           0);
  gemm1x1_kernel<<<g1024, blk, 0, stream>>>(x1, x1, 256, m2_inv_w, m2a, 1024,
                                            256, 256, 1024, nullptr,
                                            m2_inv_bias, 3);
  dwconv_kernel<3, 1><<<(4 * 1024 * HW) / 256, blk, 0, stream>>>(
      m2a, m2_dw_w, m2b, 1024, nullptr, m2_dw_bias, 3);
  gemm1x1_kernel<<<g256, blk, 0, stream>>>(m2b, m2b, 1024, m2_pw_w,
                                           (float*)d_out, 256, 1024, 1024, 256,
                                           m2_pw_g, m2_pw_b, 1);
}